// GraphAttentionNetwork_55740085567809
// MI455X (gfx1250) — compile-verified
//
#include <hip/hip_runtime.h>
#include <hip/hip_bf16.h>
#include <math.h>
#include <stdint.h>

typedef float v2f __attribute__((ext_vector_type(2)));
typedef float v8f __attribute__((ext_vector_type(8)));

#define N_NODES 100000
#define N_EDGES 1600000

__device__ __forceinline__ float lrelu(float v) { return v > 0.f ? v : 0.2f * v; }
__device__ __forceinline__ float sigmoidf(float v) { return 1.f / (1.f + __expf(-v)); }

// Monotonic float atomic-max via signed/unsigned int ordering trick.
__device__ __forceinline__ void atomicMaxF(float* addr, float val) {
  if (val >= 0.f) atomicMax((int*)addr, __float_as_int(val));
  else            atomicMin((unsigned int*)addr, __float_as_uint(val));
}

// ---------------------------------------------------------------------------
// FP32 WMMA GEMM with async-LDS B-tile staging.
// C[M,N] = A[M,K] @ B[K,N], row-major, K == 256, M%16==0, N%(16*NT)==0.
// Block = 4 waves sharing one B tile [256 x 16*NT] in LDS (async-tensor path,
// tracked by ASYNCcnt); each wave computes a 16x(16*NT) tile, 4 consecutive
// M-tiles per block.  v_wmma_f32_16x16x4_f32 inner loop fed by ds_load_2addr.
// ---------------------------------------------------------------------------
template <int NT>
__global__ void k_gemm(const float* __restrict__ A, const float* __restrict__ B,
                       float* __restrict__ C, int M, int N, int mtGroups) {
  constexpr int KK = 256;
  constexpr int TW = 16 * NT;
  __shared__ float sB[KK * TW];

  int tid = threadIdx.x;
  int ntg = blockIdx.x / mtGroups;
  int mtb = (blockIdx.x - ntg * mtGroups) * 4;  // 4 waves -> 4 M-tiles
  int n0 = ntg * TW;

  // Async-stage B tile [KK x TW] into LDS: global_load_async_to_lds_b128,
  // one 16B packet per lane per iteration, completion via ASYNCcnt.
  for (int f = tid * 4; f < KK * TW; f += blockDim.x * 4) {
    int k = f / TW;
    int c = f - k * TW;
    unsigned loff = (unsigned)(uintptr_t)(&sB[f]);  // LDS aperture: addr[31:0]
    const float* gp = B + (size_t)k * N + n0 + c;
    asm volatile("global_load_async_to_lds_b128 %0, %1, off"
                 :: "v"(loff), "v"((unsigned long long)(uintptr_t)gp)
                 : "memory");
  }
  asm volatile("s_wait_asynccnt 0" ::: "memory");
  __syncthreads();

  int lane = tid & 31;
  int mt = mtb + (tid >> 5);
  if (mt >= M / 16) return;  // wave-uniform exit: EXEC all-ones for WMMA
  int lm = lane & 15, hi = lane >> 4;
  int m0 = mt * 16;

  v8f acc[NT] = {};
  const float* Arow = A + (size_t)(m0 + lm) * KK;
  for (int k0 = 0; k0 < KK; k0 += 4) {
    int ka = k0 + 2 * hi;
    v2f a = *(const v2f*)(Arow + ka);  // A[m][ka], A[m][ka+1]
#pragma unroll
    for (int t = 0; t < NT; t++) {
      v2f b;
      b.x = sB[ka * TW + t * 16 + lm];        // B[ka][n]
      b.y = sB[(ka + 1) * TW + t * 16 + lm];  // B[ka+1][n]
      acc[t] = __builtin_amdgcn_wmma_f32_16x16x4_f32(false, a, false, b,
                                                     (short)0, acc[t], false, false);
    }
  }
#pragma unroll
  for (int t = 0; t < NT; t++) {
    float* Cp = C + (size_t)(m0 + 8 * hi) * N + n0 + t * 16 + lm;
#pragma unroll
    for (int r = 0; r < 8; r++) Cp[(size_t)r * N] = acc[t][r];
  }
}

// ---------------------------------------------------------------------------
// Per-node attention dots a_src/a_dst, init segment-max/denominator, zero the
// aggregation accumulator.
// ---------------------------------------------------------------------------
__global__ void k_att_prep(const float* __restrict__ h, const float* __restrict__ av,
                           const float* __restrict__ dv, float* __restrict__ as_,
                           float* __restrict__ ad_, float* __restrict__ m_,
                           float* __restrict__ dd_, float* __restrict__ outac,
                           int Nn, int H, int C) {
  __shared__ float sa[256], sd[256];
  int HC = H * C;
  for (int j = threadIdx.x; j < HC; j += blockDim.x) { sa[j] = av[j]; sd[j] = dv[j]; }
  __syncthreads();
  int n = blockIdx.x * blockDim.x + threadIdx.x;
  if (n >= Nn) return;
  const float* hr = h + (size_t)n * HC;
  float* oa = outac + (size_t)n * HC;
  for (int hh = 0; hh < H; hh++) {
    float s = 0.f, dsum = 0.f;
    int b0 = hh * C;
    for (int c = 0; c < C; c++) {
      float v = hr[b0 + c];
      s += v * sa[b0 + c];
      dsum += v * sd[b0 + c];
    }
    as_[(size_t)n * H + hh] = s;
    ad_[(size_t)n * H + hh] = dsum;
    m_[(size_t)n * H + hh] = -1e30f;
    dd_[(size_t)n * H + hh] = 0.f;
  }
  for (int j = 0; j < HC; j++) oa[j] = 0.f;
}

// Segment max over incoming edges (per head).
__global__ void k_edge_max(const int* __restrict__ ei, int E, const float* __restrict__ as_,
                           const float* __restrict__ ad_, float* __restrict__ m_, int H) {
  int e = blockIdx.x * blockDim.x + threadIdx.x;
  if (e >= E) return;
  int s = ei[e], d = ei[E + e];
  for (int hh = 0; hh < H; hh++) {
    float v = lrelu(as_[(size_t)s * H + hh] + ad_[(size_t)d * H + hh]);
    atomicMaxF(&m_[(size_t)d * H + hh], v);
  }
}

// Fold self-loop into max, seed denominator with self-loop exp term.
__global__ void k_node_selfterm(const float* __restrict__ as_, const float* __restrict__ ad_,
                                float* __restrict__ m_, float* __restrict__ dd_, int total) {
  int i = blockIdx.x * blockDim.x + threadIdx.x;
  if (i >= total) return;
  float es = lrelu(as_[i] + ad_[i]);
  float m = fmaxf(m_[i], es);
  m_[i] = m;
  dd_[i] = __expf(es - m);
}

// ee = exp(e - max[dst]); store per-edge, accumulate denominator.
__global__ void k_edge_expsum(const int* __restrict__ ei, int E, const float* __restrict__ as_,
                              const float* __restrict__ ad_, const float* __restrict__ m_,
                              float* __restrict__ dd_, float* __restrict__ ee_, int H) {
  int e = blockIdx.x * blockDim.x + threadIdx.x;
  if (e >= E) return;
  int s = ei[e], d = ei[E + e];
  for (int hh = 0; hh < H; hh++) {
    float v = lrelu(as_[(size_t)s * H + hh] + ad_[(size_t)d * H + hh]);
    float ee = __expf(v - m_[(size_t)d * H + hh]);
    ee_[(size_t)e * H + hh] = ee;
    atomicAdd(&dd_[(size_t)d * H + hh], ee);
  }
}

// Heavy pass: out[dst] += h[src] * alpha  (one wave per edge, lane = channel).
__global__ void k_edge_aggr(const int* __restrict__ ei, int E, const float* __restrict__ hsrc,
                            const float* __restrict__ ee_, const float* __restrict__ dd_,
                            float* __restrict__ outac, int H, int HC, int cshift) {
  int wid = (int)((blockIdx.x * blockDim.x + threadIdx.x) >> 5);
  int lane = (int)(threadIdx.x & 31);
  if (wid >= E) return;
  int s = ei[wid], d = ei[E + wid];
  float alpha[4];
  for (int hh = 0; hh < H; hh++)
    alpha[hh] = ee_[(size_t)wid * H + hh] / (dd_[(size_t)d * H + hh] + 1e-16f);
  const float* hr = hsrc + (size_t)s * HC;
  float* oa = outac + (size_t)d * HC;
  for (int ch = lane; ch < HC; ch += 32)
    atomicAdd(&oa[ch], hr[ch] * alpha[ch >> cshift]);
}

// Self-loop contribution + bias (+ optional ELU); writes final layer output.
__global__ void k_node_final(const float* __restrict__ h_, const float* __restrict__ outac,
                             const float* __restrict__ as_, const float* __restrict__ ad_,
                             const float* __restrict__ m_, const float* __restrict__ dd_,
                             const float* __restrict__ bias, float* __restrict__ dst,
                             int total, int H, int hcshift, int cshift, int elu) {
  int idx = blockIdx.x * blockDim.x + threadIdx.x;
  if (idx >= total) return;
  int n = idx >> hcshift;
  int ch = idx & ((1 << hcshift) - 1);
  int hh = ch >> cshift;
  float es = lrelu(as_[(size_t)n * H + hh] + ad_[(size_t)n * H + hh]);
  float aself = __expf(es - m_[(size_t)n * H + hh]) / (dd_[(size_t)n * H + hh] + 1e-16f);
  float v = outac[idx] + h_[idx] * aself + bias[ch];
  if (elu) v = v > 0.f ? v : (__expf(v) - 1.f);
  dst[idx] = v;
}

// node_weights: sigmoid(relu(h2 @ Wp1 + bp1) @ Wp2 + bp2), one thread per node.
__global__ void k_node_mlp(const float* __restrict__ h2, const float* __restrict__ Wp1,
                           const float* __restrict__ bp1, const float* __restrict__ Wp2,
                           const float* __restrict__ bp2, float* __restrict__ out, int Nn) {
  __shared__ float w1[32 * 64];
  __shared__ float w2[64];
  __shared__ float bb[64];
  for (int j = threadIdx.x; j < 2048; j += blockDim.x) w1[j] = Wp1[j];
  if (threadIdx.x < 64) { w2[threadIdx.x] = Wp2[threadIdx.x]; bb[threadIdx.x] = bp1[threadIdx.x]; }
  __syncthreads();
  int n = blockIdx.x * blockDim.x + threadIdx.x;
  if (n >= Nn) return;
  const float* hr = h2 + (size_t)n * 32;
  float xr[32];
  for (int k = 0; k < 32; k++) xr[k] = hr[k];
  float acc = 0.f;
  for (int j = 0; j < 64; j++) {
    float hsum = bb[j];
    for (int k = 0; k < 32; k++) hsum += xr[k] * w1[k * 64 + j];
    acc += fmaxf(hsum, 0.f) * w2[j];
  }
  out[n] = sigmoidf(acc + bp2[0]);
}

// edge_weights via WMMA: 16 edges/wave. A = concat(h2[src],h2[dst]) 16x64,
// B = Wr1 64x64 (LDS), then relu + 64->1 projection with half-wave shfl reduce.
__global__ void k_edge_mlp(const int* __restrict__ ei, int E, const float* __restrict__ h2,
                           const float* __restrict__ Wr1, const float* __restrict__ br1,
                           const float* __restrict__ Wr2, const float* __restrict__ br2,
                           float* __restrict__ out) {
  __shared__ float sW[64 * 64];
  for (int j = threadIdx.x; j < 4096; j += blockDim.x) sW[j] = Wr1[j];
  __syncthreads();
  int lane = (int)(threadIdx.x & 31);
  int e0 = (int)((blockIdx.x * (blockDim.x >> 5) + (threadIdx.x >> 5)) * 16);
  if (e0 >= E) return;  // wave-uniform
  int lm = lane & 15, hi = lane >> 4;
  int me = e0 + lm;
  const float* rs = h2 + (size_t)ei[me] * 32;      // src features
  const float* rd = h2 + (size_t)ei[E + me] * 32;  // dst features
  v8f acc[4] = {};
  for (int k0 = 0; k0 < 64; k0 += 4) {
    int ka = k0 + 2 * hi;
    const float* abase = (ka < 32) ? (rs + ka) : (rd + ka - 32);
    v2f a = *(const v2f*)abase;
#pragma unroll
    for (int t = 0; t < 4; t++) {
      v2f b;
      b.x = sW[ka * 64 + t * 16 + lm];
      b.y = sW[(ka + 1) * 64 + t * 16 + lm];
      acc[t] = __builtin_amdgcn_wmma_f32_16x16x4_f32(false, a, false, b,
                                                     (short)0, acc[t], false, false);
    }
  }
  // acc[t][r] = hidden[edge row 8*hi+r][col t*16+lm]; apply bias/relu, dot Wr2.
  float partial[8];
#pragma unroll
  for (int r = 0; r < 8; r++) {
    float p = 0.f;
#pragma unroll
    for (int t = 0; t < 4; t++) {
      int nIdx = t * 16 + lm;
      p += fmaxf(acc[t][r] + br1[nIdx], 0.f) * Wr2[nIdx];
    }
    partial[r] = p;
  }
#pragma unroll
  for (int off = 1; off < 16; off <<= 1)
#pragma unroll
    for (int r = 0; r < 8; r++) partial[r] += __shfl_xor(partial[r], off, 32);
  if (lm == 0) {
#pragma unroll
    for (int r = 0; r < 8; r++)
      out[e0 + 8 * hi + r] = sigmoidf(partial[r] + br2[0]);
  }
}

// ---------------------------------------------------------------------------
extern "C" void kernel_launch(void* const* d_in, const int* in_sizes, int n_in,
                              void* d_out, int out_size, void* d_ws, size_t ws_size,
                              hipStream_t stream) {
  (void)in_sizes; (void)n_in; (void)out_size; (void)ws_size;
  const float* x    = (const float*)d_in[0];
  const int*   ei   = (const int*)d_in[1];
  const float* W1   = (const float*)d_in[2];
  const float* asv1 = (const float*)d_in[3];
  const float* adv1 = (const float*)d_in[4];
  const float* b1   = (const float*)d_in[5];
  const float* W2   = (const float*)d_in[6];
  const float* asv2 = (const float*)d_in[7];
  const float* adv2 = (const float*)d_in[8];
  const float* b2   = (const float*)d_in[9];
  const float* Wp1  = (const float*)d_in[10];
  const float* bp1  = (const float*)d_in[11];
  const float* Wp2  = (const float*)d_in[12];
  const float* bp2  = (const float*)d_in[13];
  const float* Wr1  = (const float*)d_in[14];
  const float* br1  = (const float*)d_in[15];
  const float* Wr2  = (const float*)d_in[16];
  const float* br2  = (const float*)d_in[17];

  float* ws = (float*)d_ws;
  size_t o = 0;
  float* h1t = ws + o; o += (size_t)N_NODES * 256;  // x @ W1
  float* h1a = ws + o; o += (size_t)N_NODES * 256;  // layer-1 aggregate / elu output
  float* as1 = ws + o; o += (size_t)N_NODES * 4;
  float* ad1 = ws + o; o += (size_t)N_NODES * 4;
  float* m1  = ws + o; o += (size_t)N_NODES * 4;
  float* dd1 = ws + o; o += (size_t)N_NODES * 4;
  float* ee1 = ws + o; o += (size_t)N_EDGES * 4;
  float* h2t = ws + o; o += (size_t)N_NODES * 32;   // h1 @ W2
  float* as2 = ws + o; o += (size_t)N_NODES;
  float* ad2 = ws + o; o += (size_t)N_NODES;
  float* m2  = ws + o; o += (size_t)N_NODES;
  float* dd2 = ws + o; o += (size_t)N_NODES;
  float* ee2 = ws + o; o += (size_t)N_EDGES;

  float* h2 = (float*)d_out;                 // [N,32]
  float* nw = h2 + (size_t)N_NODES * 32;     // [N,1]
  float* ew = nw + (size_t)N_NODES;          // [E,1]

  const int TB = 256;
  int nodeBlk = (N_NODES + TB - 1) / TB;
  int tilesM = N_NODES / 16;                 // 6250
  int mtGroups = (tilesM + 3) / 4;           // 1563 (4 M-tiles per block)

  // ---- Layer 1 (H=4, C=64, HC=256) ----
  k_gemm<4><<<mtGroups * (256 / 64), 128, 0, stream>>>(x, W1, h1t, N_NODES, 256, mtGroups);
  k_att_prep<<<nodeBlk, TB, 0, stream>>>(h1t, asv1, adv1, as1, ad1, m1, dd1, h1a, N_NODES, 4, 64);
  k_edge_max<<<N_EDGES / TB, TB, 0, stream>>>(ei, N_EDGES, as1, ad1, m1, 4);
  k_node_selfterm<<<(N_NODES * 4 + TB - 1) / TB, TB, 0, stream>>>(as1, ad1, m1, dd1, N_NODES * 4);
  k_edge_expsum<<<N_EDGES / TB, TB, 0, stream>>>(ei, N_EDGES, as1, ad1, m1, dd1, ee1, 4);
  k_edge_aggr<<<N_EDGES / 8, TB, 0, stream>>>(ei, N_EDGES, h1t, ee1, dd1, h1a, 4, 256, 6);
  k_node_final<<<(N_NODES * 256) / TB, TB, 0, stream>>>(h1t, h1a, as1, ad1, m1, dd1, b1, h1a,
                                                        N_NODES * 256, 4, 8, 6, 1);

  // ---- Layer 2 (H=1, C=32, HC=32) ----
  k_gemm<2><<<mtGroups, 128, 0, stream>>>(h1a, W2, h2t, N_NODES, 32, mtGroups);
  k_att_prep<<<nodeBlk, TB, 0, stream>>>(h2t, asv2, adv2, as2, ad2, m2, dd2, h2, N_NODES, 1, 32);
  k_edge_max<<<N_EDGES / TB, TB, 0, stream>>>(ei, N_EDGES, as2, ad2, m2, 1);
  k_node_selfterm<<<nodeBlk, TB, 0, stream>>>(as2, ad2, m2, dd2, N_NODES);
  k_edge_expsum<<<N_EDGES / TB, TB, 0, stream>>>(ei, N_EDGES, as2, ad2, m2, dd2, ee2, 1);
  k_edge_aggr<<<N_EDGES / 8, TB, 0, stream>>>(ei, N_EDGES, h2t, ee2, dd2, h2, 1, 32, 5);
  k_node_final<<<(N_NODES * 32) / TB, TB, 0, stream>>>(h2t, h2, as2, ad2, m2, dd2, b2, h2,
                                                       N_NODES * 32, 1, 5, 5, 0);

  // ---- Predictor heads ----
  k_node_mlp<<<nodeBlk, TB, 0, stream>>>(h2, Wp1, bp1, Wp2, bp2, nw, N_NODES);
  k_edge_mlp<<<N_EDGES / 16 / 8, TB, 0, stream>>>(ei, N_EDGES, h2, Wr1, br1, Wr2, br2, ew);
}